// TunedGCN_8254927143331
// MI455X (gfx1250) — compile-verified
//
#include <hip/hip_runtime.h>
#include <hip/hip_bf16.h>

typedef __attribute__((ext_vector_type(2))) float v2f;
typedef __attribute__((ext_vector_type(8))) float v8f;

#define EMBED 128
#define WT_STRIDE 132                       // 132 mod 64 == 4 -> conflict-free b64 B-fragment loads
#define LDS_BYTES (WT_STRIDE * EMBED * 4)   // 67,584 B (fits easily in 320KB/WGP)

// ---------------- small helpers ----------------

__global__ void softmax_small(const float* __restrict__ alpha,
                              float* __restrict__ asoft, int n) {
  if (threadIdx.x == 0 && blockIdx.x == 0) {
    float m = alpha[0];
    for (int i = 1; i < n; ++i) m = fmaxf(m, alpha[i]);
    float s = 0.f;
    for (int i = 0; i < n; ++i) s += __expf(alpha[i] - m);
    float inv = 1.f / s;
    for (int i = 0; i < n; ++i) asoft[i] = __expf(alpha[i] - m) * inv;
  }
}

__global__ void zero_f4(float4* __restrict__ p, long n4) {
  long i = (long)blockIdx.x * blockDim.x + threadIdx.x;
  if (i < n4) p[i] = make_float4(0.f, 0.f, 0.f, 0.f);
}

// out = asoft[0] * all_emb   (layer-0 term of the weighted sum)
__global__ void scale_init(const float4* __restrict__ src,
                           const float* __restrict__ asoft,
                           float4* __restrict__ dst, long n4) {
  long i = (long)blockIdx.x * blockDim.x + threadIdx.x;
  if (i < n4) {
    float a = asoft[0];
    float4 v = src[i];
    dst[i] = make_float4(a * v.x, a * v.y, a * v.z, a * v.w);
  }
}

// ---------------- SpMM scatter: one wave per edge ----------------
// Gather h[src] row (coalesced float4 per lane), scatter-add into tmp[dst]
// with global_atomic_add_f32. h (51.2MB) is L2-resident on MI455X (192MB L2),
// so both sides of this stay in L2.
__global__ void __launch_bounds__(256)
spmm_scatter(const int* __restrict__ esrc, const int* __restrict__ edst,
             const float* __restrict__ ew, const float* __restrict__ H,
             float* __restrict__ T, long E) {
  long gid = (long)blockIdx.x * blockDim.x + threadIdx.x;
  long e = gid >> 5;                 // wave-uniform edge id
  if (e >= E) return;
  int lane = (int)(gid & 31);
  int s = esrc[e];
  int d = edst[e];
  float w = ew[e];
  const float4 hv = *(const float4*)(H + (long)s * EMBED + lane * 4);
  float* o = T + (long)d * EMBED + lane * 4;
  atomicAdd(o + 0, w * hv.x);
  atomicAdd(o + 1, w * hv.y);
  atomicAdd(o + 2, w * hv.z);
  atomicAdd(o + 3, w * hv.w);
}

// ---------------- WMMA f32 16x16x4 inner product ----------------
// A fragment: lane(l) -> row = base + (l&15); VGPR0/1 = A[row][2*hi + 0/1]
// B fragment (symmetric): lane -> col = tile*16 + (l&15); VGPR0/1 = B[2*hi + 0/1][col]
//   read from LDS holding W^T with stride WT_STRIDE -> contiguous b64 per lane.
// C/D: VGPR v holds D[v + 8*hi][l&15].
__device__ __forceinline__ void gemm_strip_wmma(const float* __restrict__ Arow,
                                                const float* __restrict__ ldsWT,
                                                int hi2, int lrow, v8f acc[8]) {
#pragma unroll 4
  for (int k = 0; k < EMBED; k += 4) {
    v2f a = *(const v2f*)(Arow + k + hi2);
#pragma unroll
    for (int t = 0; t < 8; ++t) {
      v2f b = *(const v2f*)(ldsWT + (t * 16 + lrow) * WT_STRIDE + k + hi2);
      acc[t] = __builtin_amdgcn_wmma_f32_16x16x4_f32(
          /*neg_a=*/false, a, /*neg_b=*/false, b,
          /*c_mod=*/(short)0, acc[t], /*reuse_a=*/false, /*reuse_b=*/false);
    }
  }
}

// ---------------- W_comb[l] = static_w[l] @ W_lawm (one block per layer) ----
__global__ void __launch_bounds__(256)
wcomb_wmma(const float* __restrict__ SW, const float* __restrict__ WL,
           float* __restrict__ Wc) {
  extern __shared__ float ldsWT[];
  const int tid = threadIdx.x;
  const int l = blockIdx.x;
  for (int i = tid; i < EMBED * EMBED; i += blockDim.x) {
    int r = i >> 7, c = i & 127;
    ldsWT[c * WT_STRIDE + r] = WL[i];      // transpose into LDS
  }
  __syncthreads();

  int wave = tid >> 5, lane = tid & 31;
  int lrow = lane & 15, hi = lane >> 4, hi2 = hi * 2;
  int rowBase = wave * 16;                 // 8 waves x 16 rows = 128 rows
  const float* A = SW + (long)l * EMBED * EMBED;

  v8f acc[8] = {};
  gemm_strip_wmma(A + (long)(rowBase + lrow) * EMBED, ldsWT, hi2, lrow, acc);

  float* W = Wc + (long)l * EMBED * EMBED;
#pragma unroll
  for (int v = 0; v < 8; ++v) {
    int orow = rowBase + v + hi * 8;
#pragma unroll
    for (int t = 0; t < 8; ++t)
      W[orow * EMBED + t * 16 + lrow] = acc[t][v];
  }
}

// ---------------- hbuf = tmp @ Wc ; out += a * hbuf (fused epilogue) --------
__global__ void __launch_bounds__(256)
gemm_layer_fused(const float* __restrict__ A,     // tmp [N,128]
                 const float* __restrict__ Wc,    // [128,128] row-major
                 const float* __restrict__ asoft, int aidx,
                 float* __restrict__ Hout,        // [N,128]
                 float* __restrict__ Out,         // [N,128] accumulate
                 int N) {
  extern __shared__ float ldsWT[];
  const int tid = threadIdx.x;
  for (int i = tid; i < EMBED * EMBED; i += blockDim.x) {
    int r = i >> 7, c = i & 127;
    ldsWT[c * WT_STRIDE + r] = Wc[i];
  }
  __syncthreads();

  int wave = tid >> 5, lane = tid & 31;
  int lrow = lane & 15, hi = lane >> 4, hi2 = hi * 2;
  long rowBase = (long)blockIdx.x * 128 + wave * 16;
  long row = rowBase + lrow;
  long arow = row < N ? row : (long)(N - 1);   // clamp: keep EXEC all-ones for WMMA

  v8f acc[8] = {};
  gemm_strip_wmma(A + arow * EMBED, ldsWT, hi2, lrow, acc);

  float aw = asoft[aidx];
#pragma unroll
  for (int v = 0; v < 8; ++v) {
    long orow = rowBase + v + hi * 8;
    if (orow < N) {
#pragma unroll
      for (int t = 0; t < 8; ++t) {
        long idx = orow * EMBED + t * 16 + lrow;
        float val = acc[t][v];
        Hout[idx] = val;
        Out[idx] += aw * val;
      }
    }
  }
}

// ---------------- driver ----------------

extern "C" void kernel_launch(void* const* d_in, const int* in_sizes, int n_in,
                              void* d_out, int out_size, void* d_ws, size_t ws_size,
                              hipStream_t stream) {
  const float* all_emb  = (const float*)d_in[0];
  const float* W_lawm   = (const float*)d_in[1];
  const float* static_w = (const float*)d_in[2];
  const float* alpha    = (const float*)d_in[3];
  const float* ew       = (const float*)d_in[4];
  const int*   esrc     = (const int*)d_in[5];
  const int*   edst     = (const int*)d_in[6];
  float* out = (float*)d_out;

  const int  L = in_sizes[3] - 1;                // 3 layers
  const long N = (long)in_sizes[0] / EMBED;      // 100000
  const long E = (long)in_sizes[4];              // 1.6M

  // workspace layout
  char* ws = (char*)d_ws;
  float* asoft = (float*)ws;                                         // 4 floats
  float* Wc    = (float*)(ws + 256);                                 // L*128*128
  size_t wcBytes = (size_t)L * EMBED * EMBED * sizeof(float);
  size_t off = 256 + ((wcBytes + 255) & ~(size_t)255);
  float* tmp = (float*)(ws + off);                                   // [N,128] spmm out
  off += (size_t)N * EMBED * sizeof(float);
  off = (off + 255) & ~(size_t)255;
  float* hbuf = (float*)(ws + off);                                  // [N,128] gemm out

  // 1) a = softmax(alpha)
  softmax_small<<<1, 32, 0, stream>>>(alpha, asoft, L + 1);

  // 2) W_comb[l] = static_w[l] @ W_lawm  (WMMA, one block per layer)
  wcomb_wmma<<<L, 256, LDS_BYTES, stream>>>(static_w, W_lawm, Wc);

  // 3) out = a0 * all_emb
  long n4 = N * EMBED / 4;
  int  b4 = (int)((n4 + 255) / 256);
  scale_init<<<b4, 256, 0, stream>>>((const float4*)all_emb, asoft, (float4*)out, n4);

  // 4) layers
  long sthreads = E * 32;
  int  sblocks  = (int)((sthreads + 255) / 256);
  int  gblocks  = (int)((N + 127) / 128);
  const float* hcur = all_emb;
  for (int l = 0; l < L; ++l) {
    zero_f4<<<b4, 256, 0, stream>>>((float4*)tmp, n4);
    spmm_scatter<<<sblocks, 256, 0, stream>>>(esrc, edst, ew, hcur, tmp, E);
    gemm_layer_fused<<<gblocks, 256, LDS_BYTES, stream>>>(
        tmp, Wc + (long)l * EMBED * EMBED, asoft, l + 1, hbuf, out, (int)N);
    hcur = hbuf;
  }
}